// AVWGCN_2061584302858
// MI455X (gfx1250) — compile-verified
//
#include <hip/hip_runtime.h>
#include <hip/hip_bf16.h>

// ---------------------------------------------------------------------------
// AVWGCN for MI455X (gfx1250, wave32, WMMA + async global->LDS).
// Dominant cost: four [2048x2048]x[2048x12288] fp32 GEMMs -> V_WMMA_F32_16X16X4_F32.
// ---------------------------------------------------------------------------

typedef __attribute__((ext_vector_type(2))) float v2f;
typedef __attribute__((ext_vector_type(8))) float v8f;

#define GN   2048   // nodes
#define GB   16     // batch
#define GC   64     // in channels
#define GO   64     // out channels
#define GT   12     // time steps
#define GE   10     // embedding dim

__device__ __forceinline__ v8f wmma4(v2f a, v2f b, v8f c) {
  // D = A(16x4,f32) x B(4x16,f32) + C(16x16,f32)
  return __builtin_amdgcn_wmma_f32_16x16x4_f32(
      false, a, false, b, (short)0, c, false, false);
}

// async global->LDS b32 (VDST = LDS byte offset VGPR, VADDR = 64b global addr)
__device__ __forceinline__ void async_g2l_b32(void* lds, const void* gptr) {
  unsigned loff = (unsigned)(unsigned long long)lds;  // LDS addr = generic[31:0]
  asm volatile("global_load_async_to_lds_b32 %0, %1, off"
               :: "v"(loff), "v"(gptr) : "memory");
}
__device__ __forceinline__ void async_wait0() {
  asm volatile("s_wait_asynccnt 0" ::: "memory");
}

// ---------------------------------------------------------------------------
// A = softmax(relu(emb @ emb^T), axis=1).  One workgroup per row.
// ---------------------------------------------------------------------------
__global__ __launch_bounds__(256)
void softmax_adj(const float* __restrict__ emb, float* __restrict__ A) {
  __shared__ float red[256];
  __shared__ float en[GE];
  int n = blockIdx.x;
  int t = threadIdx.x;
  if (t < GE) en[t] = emb[n * GE + t];
  __syncthreads();

  float l[8];
  float lmax = 0.0f;  // relu outputs are >= 0
#pragma unroll
  for (int j = 0; j < 8; ++j) {
    int m = j * 256 + t;
    float s = 0.0f;
#pragma unroll
    for (int d = 0; d < GE; ++d) s += en[d] * emb[m * GE + d];
    s = fmaxf(s, 0.0f);
    l[j] = s;
    lmax = fmaxf(lmax, s);
  }
  red[t] = lmax; __syncthreads();
  for (int off = 128; off > 0; off >>= 1) {
    if (t < off) red[t] = fmaxf(red[t], red[t + off]);
    __syncthreads();
  }
  float rmax = red[0];
  __syncthreads();
  float lsum = 0.0f;
#pragma unroll
  for (int j = 0; j < 8; ++j) { l[j] = __expf(l[j] - rmax); lsum += l[j]; }
  red[t] = lsum; __syncthreads();
  for (int off = 128; off > 0; off >>= 1) {
    if (t < off) red[t] += red[t + off];
    __syncthreads();
  }
  float inv = 1.0f / red[0];
#pragma unroll
  for (int j = 0; j < 8; ++j) A[n * GN + j * 256 + t] = l[j] * inv;
}

// ---------------------------------------------------------------------------
// Generic fp32 WMMA GEMM: D[m, col] = alpha * sum_k A[m,k] * B[k,col] (+ -I)
//   A element (m,k) at Ap[m*sAm + k*sAk]           (handles A and A^T views)
//   B column mapping:  bx=1 -> "x layout" [B,C,N,T]: colBase=(col/T)*N*T+col%T,
//                                k stride = T;      bx=0 -> dense: colBase=col
//   D uses the same column mapping (row stride T for x-layout, NCOLS dense).
// Block: 128x64 output tile, 256 threads = 8 waves, 2x2 WMMA tiles per wave.
// B tile staged with global_load_async_to_lds_b32 (ASYNCcnt path).
// ---------------------------------------------------------------------------
__global__ __launch_bounds__(256)
void gemm_wmma(const float* __restrict__ Ap, int sAm, int sAk,
               const float* __restrict__ Bp, int sBk, int bx,
               float* __restrict__ Dp, int K, int NCOLS,
               float alpha, float identAdd) {
  __shared__ float As[16][136];  // [k][m 0..127] + pad
  __shared__ float Bs[16][72];   // [k][col 0..63] + pad
  int m0 = blockIdx.y * 128;
  int c0 = blockIdx.x * 64;
  int t = threadIdx.x;
  int lane = t & 31;
  int wave = t >> 5;
  int wm = wave & 3;    // tM pair index (0..3)
  int wn = wave >> 2;   // tN pair index (0..1)
  int cl = lane & 15;
  int kh = (lane >> 4) * 2;

  v8f acc00 = {}, acc01 = {}, acc10 = {}, acc11 = {};

  for (int k0 = 0; k0 < K; k0 += 16) {
    // A tile: 128 m x 16 k (regular loads, strided source)
    for (int i = t; i < 2048; i += 256) {
      int m = i >> 4, k = i & 15;
      As[k][m] = Ap[(size_t)(m0 + m) * sAm + (size_t)(k0 + k) * sAk];
    }
    // B tile: 16 k x 64 cols (async global->LDS)
    for (int i = t; i < 1024; i += 256) {
      int c = i & 63, k = i >> 6;
      int gc = c0 + c;
      int cb = bx ? ((gc / GT) * (GN * GT) + (gc % GT)) : gc;
      async_g2l_b32(&Bs[k][c], Bp + (size_t)cb + (size_t)(k0 + k) * sBk);
    }
    async_wait0();
    __syncthreads();

    int mA0 = (wm * 2) * 16 + cl;
    int mA1 = mA0 + 16;
    int nB0 = (wn * 2) * 16 + cl;
    int nB1 = nB0 + 16;
#pragma unroll
    for (int kk = 0; kk < 16; kk += 4) {
      v2f a0; a0.x = As[kk + kh][mA0]; a0.y = As[kk + kh + 1][mA0];
      v2f a1; a1.x = As[kk + kh][mA1]; a1.y = As[kk + kh + 1][mA1];
      v2f b0; b0.x = Bs[kk + kh][nB0]; b0.y = Bs[kk + kh + 1][nB0];
      v2f b1; b1.x = Bs[kk + kh][nB1]; b1.y = Bs[kk + kh + 1][nB1];
      acc00 = wmma4(a0, b0, acc00);
      acc01 = wmma4(a0, b1, acc01);
      acc10 = wmma4(a1, b0, acc10);
      acc11 = wmma4(a1, b1, acc11);
    }
    __syncthreads();
  }

  int rh = (lane >> 4) * 8;
#pragma unroll
  for (int jm = 0; jm < 2; ++jm) {
#pragma unroll
    for (int jn = 0; jn < 2; ++jn) {
      v8f acc = jm ? (jn ? acc11 : acc10) : (jn ? acc01 : acc00);
      int gn = c0 + (wn * 2 + jn) * 16 + cl;
      int cb = bx ? ((gn / GT) * (GN * GT) + (gn % GT)) : gn;
#pragma unroll
      for (int r = 0; r < 8; ++r) {
        int gm = m0 + (wm * 2 + jm) * 16 + rh + r;
        float v = alpha * acc[r] + ((gm == gn) ? identAdd : 0.0f);
        Dp[bx ? ((size_t)cb + (size_t)gm * GT) : ((size_t)gm * NCOLS + gn)] = v;
      }
    }
  }
}

// ---------------------------------------------------------------------------
// x_av: one workgroup per node n.
//   W_n[k,i,o] = sum_d emb[n,d] wp[d,k,i,o]   (built in LDS, 48 KB)
//   x_av[(b,w), o] = sum_k sum_i xg_k[b,i,n,w] * W_n[k,i,o] + bias_n[o]
//   rows M = B*T = 192, cols O = 64 -> 12x4 = 48 tiles; 3x2 tiles per wave.
// Output written transposed to [B,O,N,T].
// ---------------------------------------------------------------------------
__global__ __launch_bounds__(256)
void xav_kernel(const float* __restrict__ x,   // k=0 source (identity support)
                const float* __restrict__ g1,  // k=1: A @ X
                const float* __restrict__ g2,  // k=2: S2 @ X
                const float* __restrict__ emb,
                const float* __restrict__ wp,  // [E][3][C][O]
                const float* __restrict__ bp,  // [E][O]
                float* __restrict__ out) {     // [B,O,N,T]
  __shared__ float Wn[3][64][64];   // 48 KB
  __shared__ float As[16][200];     // [k][row], row = b*T+w (192), padded
  __shared__ float biasn[64];

  int n = blockIdx.x;
  int t = threadIdx.x;

  float en[GE];
#pragma unroll
  for (int d = 0; d < GE; ++d) en[d] = emb[n * GE + d];

  for (int idx = t; idx < 3 * 64 * 64; idx += 256) {
    int k = idx >> 12;
    int i = (idx >> 6) & 63;
    int o = idx & 63;
    float s = 0.0f;
#pragma unroll
    for (int d = 0; d < GE; ++d) s += en[d] * wp[((d * 3 + k) * 64 + i) * 64 + o];
    Wn[k][i][o] = s;
  }
  if (t < 64) {
    float s = 0.0f;
#pragma unroll
    for (int d = 0; d < GE; ++d) s += en[d] * bp[d * 64 + t];
    biasn[t] = s;
  }

  int lane = t & 31;
  int wave = t >> 5;
  int wm = wave >> 1;   // 0..3 -> tM = wm*3 + {0,1,2}
  int wn = wave & 1;    // 0..1 -> tN = wn*2 + {0,1}
  int cl = lane & 15;
  int kh = (lane >> 4) * 2;

  v8f acc[3][2];
#pragma unroll
  for (int jm = 0; jm < 3; ++jm)
#pragma unroll
    for (int jn = 0; jn < 2; ++jn) { v8f z = {}; acc[jm][jn] = z; }

  for (int k = 0; k < 3; ++k) {
    const float* src = (k == 0) ? x : ((k == 1) ? g1 : g2);
    for (int kt = 0; kt < 4; ++kt) {
      __syncthreads();
      for (int idx = t; idx < 192 * 16; idx += 256) {   // A tile: 192 rows x 16 i
        int row = idx % 192;
        int kk  = idx / 192;
        int i = kt * 16 + kk;
        int b = row / GT, w = row % GT;
        As[kk][row] = src[((b * GC + i) * GN + n) * GT + w];
      }
      __syncthreads();

#pragma unroll
      for (int kk = 0; kk < 16; kk += 4) {
        v2f a[3], b[2];
#pragma unroll
        for (int jm = 0; jm < 3; ++jm) {
          int m = (wm * 3 + jm) * 16 + cl;
          a[jm].x = As[kk + kh][m];
          a[jm].y = As[kk + kh + 1][m];
        }
#pragma unroll
        for (int jn = 0; jn < 2; ++jn) {
          int o = (wn * 2 + jn) * 16 + cl;
          b[jn].x = Wn[k][kt * 16 + kk + kh][o];
          b[jn].y = Wn[k][kt * 16 + kk + kh + 1][o];
        }
#pragma unroll
        for (int jm = 0; jm < 3; ++jm)
#pragma unroll
          for (int jn = 0; jn < 2; ++jn)
            acc[jm][jn] = wmma4(a[jm], b[jn], acc[jm][jn]);
      }
    }
  }
  __syncthreads();

  int rh = (lane >> 4) * 8;
#pragma unroll
  for (int jm = 0; jm < 3; ++jm) {
#pragma unroll
    for (int jn = 0; jn < 2; ++jn) {
      int tM = wm * 3 + jm, tN = wn * 2 + jn;
      int o = tN * 16 + cl;
#pragma unroll
      for (int r = 0; r < 8; ++r) {
        int row = tM * 16 + rh + r;
        int b = row / GT, w = row % GT;
        out[((b * GO + o) * GN + n) * GT + w] = acc[jm][jn][r] + biasn[o];
      }
    }
  }
}

// ---------------------------------------------------------------------------
// x_gcn: per batch b, D[row=(n*T+w), o] = sum_{c=0..191} xg[c,row] W[c,o] + mb[o]
// xg channels: [0,64)=x, [64,128)=G3, [128,192)=G4 (all in x layout, so for a
// fixed (b,c) the (n,w) index is a contiguous offset).
// ---------------------------------------------------------------------------
__global__ __launch_bounds__(256)
void mlp_kernel(const float* __restrict__ x,
                const float* __restrict__ g3,
                const float* __restrict__ g4,
                const float* __restrict__ mw,   // [192][64]
                const float* __restrict__ mb,   // [64]
                float* __restrict__ out) {      // [B,O,N,T]
  const int NT = GN * GT;  // 24576
  __shared__ float Ws[192][64];  // 48 KB (whole mlp_w)
  __shared__ float As[16][72];
  __shared__ float bs[64];

  int b  = blockIdx.y;
  int r0 = blockIdx.x * 64;
  int t  = threadIdx.x;

  for (int idx = t; idx < 192 * 64; idx += 256) Ws[idx >> 6][idx & 63] = mw[idx];
  if (t < 64) bs[t] = mb[t];

  int lane = t & 31;
  int wave = t >> 5;
  int tM  = wave & 3;
  int tN0 = (wave >> 2) * 2;
  v8f acc0 = {}; v8f acc1 = {};

  for (int kt = 0; kt < 12; ++kt) {
    __syncthreads();
    for (int idx = t; idx < 1024; idx += 256) {   // A tile: 64 rows x 16 c
      int m = idx >> 4, k = idx & 15;
      int c = kt * 16 + k;
      const float* src = (c < 64) ? x : ((c < 128) ? g3 : g4);
      int cc = c & 63;
      As[k][m] = src[(b * GC + cc) * NT + r0 + m];
    }
    __syncthreads();

    int cl = lane & 15;
    int kh = (lane >> 4) * 2;
#pragma unroll
    for (int kk = 0; kk < 16; kk += 4) {
      v2f a;  a.x  = As[kk + kh][tM * 16 + cl];   a.y  = As[kk + kh + 1][tM * 16 + cl];
      v2f b0; b0.x = Ws[kt * 16 + kk + kh][tN0 * 16 + cl];
              b0.y = Ws[kt * 16 + kk + kh + 1][tN0 * 16 + cl];
      v2f b1; b1.x = Ws[kt * 16 + kk + kh][tN0 * 16 + 16 + cl];
              b1.y = Ws[kt * 16 + kk + kh + 1][tN0 * 16 + 16 + cl];
      acc0 = wmma4(a, b0, acc0);
      acc1 = wmma4(a, b1, acc1);
    }
  }
  __syncthreads();

  int cl = lane & 15;
  int rh = (lane >> 4) * 8;
#pragma unroll
  for (int j = 0; j < 2; ++j) {
    v8f acc = j ? acc1 : acc0;
    int o = (tN0 + j) * 16 + cl;
#pragma unroll
    for (int r = 0; r < 8; ++r) {
      int row = r0 + tM * 16 + rh + r;
      out[(b * GO + o) * NT + row] = acc[r] + bs[o];
    }
  }
}

// ---------------------------------------------------------------------------
extern "C" void kernel_launch(void* const* d_in, const int* in_sizes, int n_in,
                              void* d_out, int out_size, void* d_ws, size_t ws_size,
                              hipStream_t stream) {
  (void)in_sizes; (void)n_in; (void)out_size; (void)ws_size;
  const float* x   = (const float*)d_in[0];   // [B,C,N,T]
  const float* emb = (const float*)d_in[1];   // [N,E]
  const float* sup = (const float*)d_in[2];   // [1,N,N]
  const float* wp  = (const float*)d_in[3];   // [E,3,C,O]
  const float* bp  = (const float*)d_in[4];   // [E,O]
  const float* mw  = (const float*)d_in[5];   // [192,O]
  const float* mb  = (const float*)d_in[6];   // [O]
  float* out = (float*)d_out;                 // x_av then x_gcn, each [B,O,N,T]

  const size_t NN  = (size_t)GN * GN;               // 4,194,304
  const size_t XSZ = (size_t)GB * GC * GN * GT;     // 25,165,824
  float* wsf  = (float*)d_ws;
  float* Aadj = wsf;
  float* S2   = Aadj + NN;
  float* G1   = S2 + NN;
  float* G2   = G1 + XSZ;
  float* G3   = G2 + XSZ;
  float* G4   = G3 + XSZ;

  const int BCT = GB * GC * GT;  // 12288

  // 1. adaptive adjacency
  softmax_adj<<<GN, 256, 0, stream>>>(emb, Aadj);
  // 2. S2 = 2*A@A - I       (dense square GEMM)
  gemm_wmma<<<dim3(GN / 64, GN / 128), 256, 0, stream>>>(
      Aadj, GN, 1, Aadj, GN, 0, S2, GN, GN, 2.0f, -1.0f);
  // 3. G1 = A @ X           (cheb k=1)
  gemm_wmma<<<dim3(BCT / 64, GN / 128), 256, 0, stream>>>(
      Aadj, GN, 1, x, GT, 1, G1, GN, BCT, 1.0f, 0.0f);
  // 4. G2 = S2 @ X          (cheb k=2)
  gemm_wmma<<<dim3(BCT / 64, GN / 128), 256, 0, stream>>>(
      S2, GN, 1, x, GT, 1, G2, GN, BCT, 1.0f, 0.0f);
  // 5. G3 = sup^T @ X       (diffusion order 1)
  gemm_wmma<<<dim3(BCT / 64, GN / 128), 256, 0, stream>>>(
      sup, 1, GN, x, GT, 1, G3, GN, BCT, 1.0f, 0.0f);
  // 6. G4 = sup^T @ G3      (diffusion order 2)
  gemm_wmma<<<dim3(BCT / 64, GN / 128), 256, 0, stream>>>(
      sup, 1, GN, G3, GT, 1, G4, GN, BCT, 1.0f, 0.0f);
  // 7. x_av (per-node adaptive weights + bias, transposed output)
  xav_kernel<<<GN, 256, 0, stream>>>(x, G1, G2, emb, wp, bp, out);
  // 8. x_gcn (1x1 conv over concat channels + bias)
  mlp_kernel<<<dim3((GN * GT) / 64, GB), 256, 0, stream>>>(
      x, G3, G4, mw, mb, out + XSZ);
}